// NoisyTopkRouter_63067299774600
// MI455X (gfx1250) — compile-verified
//
#include <hip/hip_runtime.h>
#include <math.h>

#define BB 4
#define SS 4096
#define DD 2048
#define EE 64
#define TOPK 8
#define ROWS_PER_BLOCK 128
#define ROWS_PER_WAVE 16
#define KCHUNK 64
#define XSTRIDE 68            // LDS row stride for x tile (floats): 16B-aligned rows, conflict-free
#define WT_STRIDE 68          // LDS row stride for transposed W (floats): bank = (4n+k)%64, conflict-free
#define WT_MAT (EE * WT_STRIDE)   // floats per transposed weight matrix

typedef __attribute__((ext_vector_type(2))) float v2f;
typedef __attribute__((ext_vector_type(8))) float v8f;

__global__ __launch_bounds__(256) void noisy_topk_router_kernel(
    const float* __restrict__ x, const float* __restrict__ noise,
    const float* __restrict__ Wr, const float* __restrict__ br,
    const float* __restrict__ Wn, const float* __restrict__ bn,
    float* __restrict__ out_router, int* __restrict__ out_idx)
{
    __shared__ float lds_x[ROWS_PER_BLOCK * XSTRIDE];   // 34.0 KB: x chunk, later noisy logits
    __shared__ float lds_w[2 * WT_MAT];                 // 34.0 KB: W_route / W_noise, [expert][k] transposed

    const int tid  = threadIdx.x;
    const int wave = tid >> 5;
    const int lane = tid & 31;
    const int half = lane >> 4;     // which 16-lane half of the wave
    const int ln   = lane & 15;
    const int row0 = wave * ROWS_PER_WAVE;                         // wave's row base within block tile
    const long blk_row = (long)blockIdx.x * ROWS_PER_BLOCK;        // block's global token row base
    const long g0 = blk_row + row0;                                // wave's global token row base

    v8f acc_r[4], acc_n[4];
#pragma unroll
    for (int t = 0; t < 4; ++t) { acc_r[t] = (v8f)0.0f; acc_n[t] = (v8f)0.0f; }

    for (int k0 = 0; k0 < DD; k0 += KCHUNK) {
        __syncthreads();  // previous chunk fully consumed before restaging

        // ---- stage x chunk: 128 rows x 64 K as float4 (2048 float4 / 256 threads = 8 each)
#pragma unroll
        for (int i = 0; i < 8; ++i) {
            int e4 = tid + i * 256;
            int r  = e4 >> 4;
            int c4 = e4 & 15;
            float4 v = *reinterpret_cast<const float4*>(&x[(blk_row + r) * DD + k0 + c4 * 4]);
            *reinterpret_cast<float4*>(&lds_x[r * XSTRIDE + c4 * 4]) = v;
        }
        // ---- stage both weight chunks TRANSPOSED: global [k][n] -> LDS [n][k]
        // so that the WMMA B pair (k, k+1) at fixed expert n is one contiguous ds_load_b64
#pragma unroll
        for (int i = 0; i < 4; ++i) {
            int e4 = tid + i * 256;          // 0..1023
            int r  = e4 >> 4;                // k-row 0..63
            int c4 = e4 & 15;                // expert group 0..15
            float4 vr = *reinterpret_cast<const float4*>(&Wr[(k0 + r) * EE + c4 * 4]);
            float4 vn = *reinterpret_cast<const float4*>(&Wn[(k0 + r) * EE + c4 * 4]);
            lds_w[(c4 * 4 + 0) * WT_STRIDE + r] = vr.x;
            lds_w[(c4 * 4 + 1) * WT_STRIDE + r] = vr.y;
            lds_w[(c4 * 4 + 2) * WT_STRIDE + r] = vr.z;
            lds_w[(c4 * 4 + 3) * WT_STRIDE + r] = vr.w;
            lds_w[WT_MAT + (c4 * 4 + 0) * WT_STRIDE + r] = vn.x;
            lds_w[WT_MAT + (c4 * 4 + 1) * WT_STRIDE + r] = vn.y;
            lds_w[WT_MAT + (c4 * 4 + 2) * WT_STRIDE + r] = vn.z;
            lds_w[WT_MAT + (c4 * 4 + 3) * WT_STRIDE + r] = vn.w;
        }
        if (k0 + KCHUNK < DD) {   // emits global_prefetch_b8
            __builtin_prefetch(&Wr[(k0 + KCHUNK) * EE], 0, 0);
            __builtin_prefetch(&Wn[(k0 + KCHUNK) * EE], 0, 0);
            __builtin_prefetch(&x[(blk_row + row0 + ln) * DD + k0 + KCHUNK], 0, 0);
        }
        __syncthreads();

        // ---- 16 WMMA K-steps of 4; A layout per ISA: VGPR0 = K{0|2}, VGPR1 = K{1|3} by lane half
#pragma unroll
        for (int kk = 0; kk < KCHUNK; kk += 4) {
            const int kofs = kk + 2 * half;
            v2f a = *reinterpret_cast<const v2f*>(
                &lds_x[(row0 + ln) * XSTRIDE + kofs]);            // ds_load_b64
#pragma unroll
            for (int t = 0; t < 4; ++t) {
                const int nb = t * 16 + ln;
                v2f bR = *reinterpret_cast<const v2f*>(&lds_w[nb * WT_STRIDE + kofs]);           // ds_load_b64
                v2f bN = *reinterpret_cast<const v2f*>(&lds_w[WT_MAT + nb * WT_STRIDE + kofs]);  // ds_load_b64
                acc_r[t] = __builtin_amdgcn_wmma_f32_16x16x4_f32(
                    false, a, false, bR, (short)0, acc_r[t], false, false);
                acc_n[t] = __builtin_amdgcn_wmma_f32_16x16x4_f32(
                    false, a, false, bN, (short)0, acc_n[t], false, false);
            }
        }
    }

    // ---- epilogue: bias + noise * softplus(noise_logits) -> wave-private LDS rows
    // C/D layout: VGPR j holds M=j (lanes 0-15) / M=j+8 (lanes 16-31), N = lane%16 per N-tile
#pragma unroll
    for (int t = 0; t < 4; ++t) {
        const int nb = t * 16 + ln;
        const float biasR = br[nb];
        const float biasN = bn[nb];
#pragma unroll
        for (int j = 0; j < 8; ++j) {
            const int  ml   = j + 8 * half;
            const long grow = g0 + ml;
            float lr = acc_r[t][j] + biasR;
            float lz = acc_n[t][j] + biasN;
            float sp = (lz > 0.0f) ? (lz + log1pf(expf(-lz))) : log1pf(expf(lz));  // stable softplus
            float nz = noise[grow * EE + nb];
            lds_x[(row0 + ml) * XSTRIDE + nb] = lr + nz * sp;
        }
    }
    __syncthreads();  // cross-lane visibility of noisy logits

    // ---- top-8 + softmax + scatter, one token row per lane (lanes 0-15)
    if (lane < 16) {
        const int  r    = row0 + lane;
        const long grow = g0 + lane;
        const float* rowp = &lds_x[r * XSTRIDE];
        unsigned long long used = 0ull;
        float vals[TOPK];
        int   idx[TOPK];
#pragma unroll
        for (int k = 0; k < TOPK; ++k) {
            float best = -3.4e38f;
            int   bi   = 0;
#pragma unroll 8
            for (int e = 0; e < EE; ++e) {
                float v = rowp[e];
                if (!((used >> e) & 1ull) && v > best) { best = v; bi = e; }  // strict > : lowest-index ties, matches jax
            }
            used |= (1ull << bi);
            vals[k] = best;
            idx[k]  = bi;
        }
        float mx = vals[0];   // vals sorted descending
        float p[TOPK];
        float sum = 0.0f;
#pragma unroll
        for (int k = 0; k < TOPK; ++k) { p[k] = expf(vals[k] - mx); sum += p[k]; }
        float inv = 1.0f / sum;
#pragma unroll
        for (int k = 0; k < TOPK; ++k) out_idx[grow * TOPK + k] = idx[k];
        const float4 z4 = make_float4(0.f, 0.f, 0.f, 0.f);
#pragma unroll
        for (int e4 = 0; e4 < 16; ++e4)
            *reinterpret_cast<float4*>(&lds_x[r * XSTRIDE + e4 * 4]) = z4;   // ds_store_b128
#pragma unroll
        for (int k = 0; k < TOPK; ++k) lds_x[r * XSTRIDE + idx[k]] = p[k] * inv;
    }
    __syncthreads();

    // ---- coalesced writeback of the wave's 16x64 router-output tile (float4)
    float* orow = out_router + g0 * EE;
#pragma unroll
    for (int w = 0; w < 8; ++w) {
        int g4 = lane + w * 32;
        int g  = g4 * 4;
        int r  = g >> 6;
        int c  = g & 63;
        float4 v = *reinterpret_cast<const float4*>(&lds_x[(row0 + r) * XSTRIDE + c]);
        *reinterpret_cast<float4*>(&orow[g]) = v;
    }
}

extern "C" void kernel_launch(void* const* d_in, const int* in_sizes, int n_in,
                              void* d_out, int out_size, void* d_ws, size_t ws_size,
                              hipStream_t stream) {
    const float* x     = (const float*)d_in[0];
    const float* noise = (const float*)d_in[1];
    const float* Wr    = (const float*)d_in[2];
    const float* br    = (const float*)d_in[3];
    const float* Wn    = (const float*)d_in[4];
    const float* bn    = (const float*)d_in[5];
    float* out_router = (float*)d_out;
    int*   out_idx    = (int*)((float*)d_out + (size_t)BB * SS * EE);  // tuple outputs concatenated

    dim3 grid((BB * SS) / ROWS_PER_BLOCK);   // 128 blocks x 128 tokens
    dim3 block(256);                          // 8 wave32 waves
    noisy_topk_router_kernel<<<grid, block, 0, stream>>>(
        x, noise, Wr, br, Wn, bn, out_router, out_idx);
}